// Attention_28415503630627
// MI455X (gfx1250) — compile-verified
//
#include <hip/hip_runtime.h>

typedef float v2f __attribute__((ext_vector_type(2)));
typedef float v8f __attribute__((ext_vector_type(8)));

#define HEADS 8
#define DH 32
#define NTOK 2048
#define DIM 256
#define BATCH 2
#define SCALE2 (1.0f / 32.0f) /* (d^-0.5)^2, d = 32 */

// workspace layout (floats): Qa | Ka | Qs | Ks, each [B,H,N,DH]
#define PROJ_ELEMS ((size_t)BATCH * HEADS * NTOK * DH) /* 1,048,576 */
#define OFF_QA ((size_t)0)
#define OFF_KA (PROJ_ELEMS)
#define OFF_QS (2 * PROJ_ELEMS)
#define OFF_KS (3 * PROJ_ELEMS)

static __device__ __forceinline__ v8f wmma_f32(v2f a, v2f b, v8f c) {
  // V_WMMA_F32_16X16X4_F32: D = A(16x4) * B(4x16) + C(16x16), all f32
  return __builtin_amdgcn_wmma_f32_16x16x4_f32(
      /*neg_a=*/false, a, /*neg_b=*/false, b,
      /*c_mod=*/(short)0, c, /*reuse_a=*/false, /*reuse_b=*/false);
}

// ---------------------------------------------------------------------------
// Kernel 1: projection GEMM.  Y = X @ W for Q|K column block (cols 0..511).
// One wave computes one 16x16 output tile, K=256 in 64 WMMA steps.
// Output written in [b,h,n,dh] layout into workspace.
// grid = (4, 256, 2): x*8+wave = col tile 0..31, y = row tile over B*N, z = src
// ---------------------------------------------------------------------------
__global__ __launch_bounds__(256) void proj_kernel(
    const float* __restrict__ x, const float* __restrict__ attn_feat,
    const float* __restrict__ Wq, const float* __restrict__ Waux,
    float* __restrict__ ws) {
  const int lane = threadIdx.x & 31;
  const int wave = threadIdx.x >> 5;
  const int hi = lane >> 4;      // K-half select
  const int lo = lane & 15;
  const int ct = blockIdx.x * 8 + wave;  // col tile 0..31 (cols 0..511)
  const int m0 = blockIdx.y * 16;        // row tile over B*N = 4096
  const int src = blockIdx.z;            // 0 = self (x,Wq), 1 = aux
  const float* X = src ? attn_feat : x;
  const float* W = src ? Waux : Wq;
  float* Qdst = ws + (src ? OFF_QA : OFF_QS);
  float* Kdst = ws + (src ? OFF_KA : OFF_KS);

  const int c = ct * 16 + lo;  // output column 0..511
  const float* Arow = X + (size_t)(m0 + lo) * DIM;

  v8f acc = {0.f, 0.f, 0.f, 0.f, 0.f, 0.f, 0.f, 0.f};
#pragma unroll 8
  for (int s = 0; s < 64; ++s) {
    const int k = s * 4 + hi * 2;
    v2f a = *(const v2f*)(Arow + k);  // A[m][k], A[m][k+1]
    v2f b;
    b.x = W[(size_t)k * 768 + c];        // B[k][n]
    b.y = W[(size_t)(k + 1) * 768 + c];  // B[k+1][n]
    acc = wmma_f32(a, b, acc);
  }

  float* dst;
  int hh;
  if (c < 256) { dst = Qdst; hh = c >> 5; }
  else         { dst = Kdst; hh = (c - 256) >> 5; }
  const int dh = c & 31;
#pragma unroll
  for (int r = 0; r < 8; ++r) {
    const int row = m0 + r + hi * 8;  // global row over B*N
    const int b_ = row >> 11, nn = row & 2047;
    dst[(((size_t)(b_ * HEADS + hh)) * NTOK + nn) * DH + dh] = acc[r];
  }
}

// ---------------------------------------------------------------------------
// Kernel 2: fused dual-score flash attention.
// Block = 256 threads = 8 waves = 4 query tiles (16 rows) x 2 dim-halves.
// Per 16-key block: stage x tile + K tiles into LDS, each wave computes
// S_aux, S_self (8+8 WMMA), P = S_aux*S_self*SCALE^2, online softmax,
// LDS transpose of P to A-layout, then O += P @ x (32 WMMA).
// grid = (N/64 = 32, B*H = 16)
// ---------------------------------------------------------------------------
__global__ __launch_bounds__(256) void attn_kernel(
    const float* __restrict__ x, const float* __restrict__ ws,
    float* __restrict__ out0) {
  __shared__ float sX[16 * 272];      // x key-block, stride 272 (bank-safe)
  __shared__ float sKa[16 * 36];      // K_aux block, stride 36
  __shared__ float sKs[16 * 36];      // K_self block
  __shared__ float sP[8 * 16 * 18];   // per-wave P transpose scratch

  const int tid = threadIdx.x;
  const int lane = tid & 31, wave = tid >> 5;
  const int half = wave & 1, qt = wave >> 1;
  const int hi = lane >> 4, lo = lane & 15;
  const int bh = blockIdx.y, b = bh >> 3, h = bh & 7;
  const int m0 = blockIdx.x * 64 + qt * 16;  // query rows of this wave

  const float* Qa = ws + OFF_QA;
  const float* Qs = ws + OFF_QS;
  const float* kabase = ws + OFF_KA + (size_t)bh * NTOK * DH;
  const float* ksbase = ws + OFF_KS + (size_t)bh * NTOK * DH;
  const float* xb = x + (size_t)b * NTOK * DIM;
  float* myP = sP + wave * (16 * 18);

  // Preload Q operands in WMMA A-layout: 8 K-steps of v2f.
  v2f qa[8], qs[8];
  {
    const float* qab = Qa + ((size_t)bh * NTOK + m0 + lo) * DH;
    const float* qsb = Qs + ((size_t)bh * NTOK + m0 + lo) * DH;
#pragma unroll
    for (int kk = 0; kk < 8; ++kk) {
      const int k = kk * 4 + hi * 2;
      qa[kk] = *(const v2f*)(qab + k);
      qs[kk] = *(const v2f*)(qsb + k);
    }
  }

  v8f O[8];
#pragma unroll
  for (int t = 0; t < 8; ++t) O[t] = (v8f){0.f, 0.f, 0.f, 0.f, 0.f, 0.f, 0.f, 0.f};
  float m[8], l[8];
#pragma unroll
  for (int r = 0; r < 8; ++r) { m[r] = -3.0e38f; l[r] = 0.f; }

  for (int j0 = 0; j0 < NTOK; j0 += 16) {
    __syncthreads();
    // ---- cooperative staging ----
#pragma unroll
    for (int t = 0; t < 4; ++t) {  // x block: 16 rows x 256 = 1024 float4
      const int idx = t * 256 + tid;
      const int row = idx >> 6, c4 = (idx & 63) << 2;
      *(float4*)&sX[row * 272 + c4] =
          *(const float4*)(xb + (size_t)(j0 + row) * DIM + c4);
    }
    if (tid < 128) {  // K_aux: 16 rows x 32 = 128 float4
      const int row = tid >> 3, c4 = (tid & 7) << 2;
      *(float4*)&sKa[row * 36 + c4] =
          *(const float4*)(kabase + (size_t)(j0 + row) * DH + c4);
    } else {          // K_self
      const int t2 = tid - 128;
      const int row = t2 >> 3, c4 = (t2 & 7) << 2;
      *(float4*)&sKs[row * 36 + c4] =
          *(const float4*)(ksbase + (size_t)(j0 + row) * DH + c4);
    }
    __syncthreads();

    // ---- S_aux = Qa Ka^T, S_self = Qs Ks^T  (B[k][n] = K[n][k]) ----
    v8f Sa = {0.f, 0.f, 0.f, 0.f, 0.f, 0.f, 0.f, 0.f};
    v8f Ss = {0.f, 0.f, 0.f, 0.f, 0.f, 0.f, 0.f, 0.f};
#pragma unroll
    for (int kk = 0; kk < 8; ++kk) {
      const int k = kk * 4 + hi * 2;
      v2f ba = *(const v2f*)&sKa[lo * 36 + k];
      Sa = wmma_f32(qa[kk], ba, Sa);
      v2f bs = *(const v2f*)&sKs[lo * 36 + k];
      Ss = wmma_f32(qs[kk], bs, Ss);
    }

    float P[8];
#pragma unroll
    for (int r = 0; r < 8; ++r) P[r] = Sa[r] * Ss[r] * SCALE2;

    // ---- online softmax: C-layout rows live across 16-lane halves ----
    float alpha[8];
#pragma unroll
    for (int r = 0; r < 8; ++r) {
      float v = P[r];
      v = fmaxf(v, __shfl_xor(v, 1));
      v = fmaxf(v, __shfl_xor(v, 2));
      v = fmaxf(v, __shfl_xor(v, 4));
      v = fmaxf(v, __shfl_xor(v, 8));
      const float mn = fmaxf(m[r], v);
      alpha[r] = __expf(m[r] - mn);
      m[r] = mn;
    }
#pragma unroll
    for (int r = 0; r < 8; ++r) {
      float e = __expf(P[r] - m[r]);
      P[r] = e;
      float s = e;
      s += __shfl_xor(s, 1);
      s += __shfl_xor(s, 2);
      s += __shfl_xor(s, 4);
      s += __shfl_xor(s, 8);
      l[r] = l[r] * alpha[r] + s;
    }
#pragma unroll
    for (int t = 0; t < 8; ++t)
#pragma unroll
      for (int r = 0; r < 8; ++r) O[t][r] *= alpha[r];

    // ---- transpose P (C-layout -> A-layout) via padded LDS scratch ----
#pragma unroll
    for (int r = 0; r < 8; ++r) myP[(r + hi * 8) * 18 + lo] = P[r];
    asm volatile("s_wait_dscnt 0" ::: "memory");
    v2f ap[4];
#pragma unroll
    for (int kk = 0; kk < 4; ++kk)
      ap[kk] = *(const v2f*)&myP[lo * 18 + kk * 4 + hi * 2];

    // ---- O += P(16x16) @ X(16x128 half): 8 col tiles x 4 K-steps ----
#pragma unroll
    for (int t = 0; t < 8; ++t) {
      const int n0 = half * 128 + t * 16 + lo;
      v8f o = O[t];
#pragma unroll
      for (int kk = 0; kk < 4; ++kk) {
        const int k = kk * 4 + hi * 2;
        v2f bx;
        bx.x = sX[k * 272 + n0];
        bx.y = sX[(k + 1) * 272 + n0];
        o = wmma_f32(ap[kk], bx, o);
      }
      O[t] = o;
    }
  }

  // ---- finalize: divide by l, store out[b][n][h*256 + col] ----
  float invl[8];
#pragma unroll
  for (int r = 0; r < 8; ++r) invl[r] = 1.0f / l[r];
#pragma unroll
  for (int t = 0; t < 8; ++t) {
    const int col = h * 256 + half * 128 + t * 16 + lo;
#pragma unroll
    for (int r = 0; r < 8; ++r) {
      const int n = m0 + r + hi * 8;
      out0[((size_t)b * NTOK + n) * 2048 + col] = O[t][r] * invl[r];
    }
  }
}

// ---------------------------------------------------------------------------
// Kernel 3: recompute attn row 0 per (b,h) for output #3 (4 MFLOP), and copy
// out[:,0] for output #2.  grid = 16 blocks (bh), 256 threads.
// ---------------------------------------------------------------------------
__global__ __launch_bounds__(256) void row0_kernel(
    const float* __restrict__ ws, const float* __restrict__ out0,
    float* __restrict__ out1, float* __restrict__ attn0) {
  __shared__ float sp[NTOK];
  __shared__ float red[256];
  __shared__ float qa_s[DH], qs_s[DH];
  __shared__ float stat[2];
  const int bh = blockIdx.x, tid = threadIdx.x;
  const float* Qa = ws + OFF_QA + (size_t)bh * NTOK * DH;
  const float* Ka = ws + OFF_KA + (size_t)bh * NTOK * DH;
  const float* Qs = ws + OFF_QS + (size_t)bh * NTOK * DH;
  const float* Ks = ws + OFF_KS + (size_t)bh * NTOK * DH;
  if (tid < DH) qa_s[tid] = Qa[tid];
  else if (tid < 2 * DH) qs_s[tid - DH] = Qs[tid - DH];
  __syncthreads();

  float lmax = -3.0e38f;
  for (int j = tid; j < NTOK; j += 256) {
    float sa = 0.f, ss = 0.f;
    const float* ka = Ka + (size_t)j * DH;
    const float* ks = Ks + (size_t)j * DH;
#pragma unroll
    for (int d = 0; d < DH; ++d) { sa += qa_s[d] * ka[d]; ss += qs_s[d] * ks[d]; }
    const float p = sa * ss * SCALE2;
    sp[j] = p;
    lmax = fmaxf(lmax, p);
  }
  red[tid] = lmax;
  __syncthreads();
  for (int s = 128; s > 0; s >>= 1) {
    if (tid < s) red[tid] = fmaxf(red[tid], red[tid + s]);
    __syncthreads();
  }
  if (tid == 0) stat[0] = red[0];
  __syncthreads();
  const float mmax = stat[0];
  float lsum = 0.f;
  for (int j = tid; j < NTOK; j += 256) {
    const float e = __expf(sp[j] - mmax);
    sp[j] = e;
    lsum += e;
  }
  red[tid] = lsum;
  __syncthreads();
  for (int s = 128; s > 0; s >>= 1) {
    if (tid < s) red[tid] += red[tid + s];
    __syncthreads();
  }
  if (tid == 0) stat[1] = red[0];
  __syncthreads();
  const float inv = 1.0f / stat[1];
  for (int j = tid; j < NTOK; j += 256)
    attn0[(size_t)bh * NTOK + j] = sp[j] * inv;

  if (bh < BATCH) {  // out[:,0] = out[b][0][:]
    for (int c = tid; c < 2048; c += 256)
      out1[bh * 2048 + c] = out0[(size_t)bh * 2048 * 2048 + c];
  }
}

// ---------------------------------------------------------------------------
extern "C" void kernel_launch(void* const* d_in, const int* in_sizes, int n_in,
                              void* d_out, int out_size, void* d_ws,
                              size_t ws_size, hipStream_t stream) {
  const float* x  = (const float*)d_in[0];
  const float* af = (const float*)d_in[1];
  const float* Wq = (const float*)d_in[2];
  const float* Wa = (const float*)d_in[3];
  // d_in[4] (use_cls_tokens) is unused by the math paths that survive.
  float* ws = (float*)d_ws;  // needs 4 * 1,048,576 floats = 16 MB
  float* out0 = (float*)d_out;                         // [2,2048,2048]
  float* out1 = out0 + (size_t)BATCH * NTOK * 2048;    // [2,2048]
  float* attn0 = out1 + (size_t)BATCH * 2048;          // [2,8*2048]

  proj_kernel<<<dim3(4, 256, 2), 256, 0, stream>>>(x, af, Wq, Wa, ws);
  attn_kernel<<<dim3(32, 16), 256, 0, stream>>>(x, ws, out0);
  row0_kernel<<<16, 256, 0, stream>>>(ws, out0, out1, attn0);
}